// GNNTumorSegmentation_78005196030173
// MI455X (gfx1250) — compile-verified
//
#include <hip/hip_runtime.h>
#include <hip/hip_bf16.h>

typedef __attribute__((ext_vector_type(2))) float v2f;
typedef __attribute__((ext_vector_type(8))) float v8f;

// ---------------------------------------------------------------------------
// Edge preprocessing
// ---------------------------------------------------------------------------
__global__ __launch_bounds__(256)
void ew_deg_kernel(const float* __restrict__ edge_attr, const int* __restrict__ ei,
                   const float* __restrict__ We, const float* __restrict__ be,
                   float* __restrict__ ew, float* __restrict__ deg, int E) {
    int e = blockIdx.x * blockDim.x + threadIdx.x;
    if (e >= E) return;
    float w = edge_attr[e] * We[0] + be[0];
    ew[e] = w;
    atomicAdd(&deg[ei[E + e]], w);   // dst row of edge_index
}

__global__ __launch_bounds__(256)
void dinv_kernel(const float* __restrict__ deg, float* __restrict__ dinv,
                 float* __restrict__ selfn, int N) {
    int n = blockIdx.x * blockDim.x + threadIdx.x;
    if (n >= N) return;
    float d = deg[n] + 1.0f;           // self-loop weight 1.0
    float di = rsqrtf(d);
    dinv[n] = di;
    selfn[n] = di * di;
}

__global__ __launch_bounds__(256)
void norm_kernel(const int* __restrict__ ei, const float* __restrict__ ew,
                 const float* __restrict__ dinv, float* __restrict__ nrm, int E) {
    int e = blockIdx.x * blockDim.x + threadIdx.x;
    if (e >= E) return;
    nrm[e] = dinv[ei[e]] * ew[e] * dinv[ei[E + e]];
}

// ---------------------------------------------------------------------------
// WMMA GEMM: C[N x ncols] = (opt relu)(A[N x K] @ B[K x ncols] + bias)
// One wave per 16x16 f32 tile via V_WMMA_F32_16X16X4_F32, K stepped by 4.
//
// B is staged into LDS with GLOBAL_LOAD_ASYNC_TO_LDS_B32 (ASYNCcnt path,
// no VGPR round-trip), pre-swizzled into a pair-packed layout so that the
// two B values a lane needs per K-step -- B[k0+half][c] and B[k0+half+2][c]
// -- are adjacent: one ds_load_b64 per K-step instead of two ds_load_b32.
// A-operand pairs are contiguous in memory -> v2f (global_load_b64).
// EXEC stays all-1s through every WMMA (clamped loads, guarded stores only
// after the MMA loop).
// ---------------------------------------------------------------------------
__global__ __launch_bounds__(256)
void gemm_wmma_kernel(const float* __restrict__ A, const float* __restrict__ B,
                      const float* __restrict__ bias, float* __restrict__ C,
                      int N, int K, int ncols, int doRelu) {
    extern __shared__ float lds[];               // K*ncols (swizzled B) + ncols (bias)
    const int tid = threadIdx.x;
    const int nB = K * ncols;
    const int cshift = (ncols == 64) ? 6 : 5;
    const int cmask = ncols - 1;

    // async-stage B into LDS, swizzled: element (k, c), k = 4q + r, lands at
    // pair index ((2q + (r&1)) * ncols + c) * 2 + (r>>1)
    for (int idx = tid; idx < nB; idx += (int)blockDim.x) {
        const int k = idx >> cshift;
        const int c = idx & cmask;
        const int q = k >> 2;
        const int r = k & 3;
        const int didx = (((q << 1) + (r & 1)) * ncols + c) * 2 + (r >> 1);
        const unsigned ldsaddr = (unsigned)(unsigned long long)(lds + didx);
        const unsigned goff = (unsigned)idx * 4u;
        asm volatile("global_load_async_to_lds_b32 %0, %1, %2"
                     : : "v"(ldsaddr), "v"(goff), "s"(B) : "memory");
    }
    float* ldsBias = lds + nB;
    if (bias) { for (int i = tid; i < ncols; i += blockDim.x) ldsBias[i] = bias[i]; }
    else      { for (int i = tid; i < ncols; i += blockDim.x) ldsBias[i] = 0.0f; }
    asm volatile("s_wait_asynccnt 0x0" ::: "memory");
    __syncthreads();

    const int wave      = tid >> 5;
    const int lane      = tid & 31;
    const int nColTiles = ncols >> 4;                       // 4 (H=64) or 2 (H/2=32)
    const int rowTile   = wave / nColTiles;
    const int colTile   = wave % nColTiles;
    const int rowsPerBlk = (((int)blockDim.x >> 5) / nColTiles) << 4;  // 32 or 64
    const long long row0 = (long long)blockIdx.x * rowsPerBlk + (long long)rowTile * 16;
    const int col0 = colTile << 4;

    const int m    = lane & 15;
    const int half = lane >> 4;                             // 0: K=0..1, 1: K=2..3 (A layout)
    long long arow = row0 + m;
    if (arow >= N) arow = N - 1;                            // clamp: keep EXEC all-1s
    const float* Arow = A + arow * (long long)K;

    v8f acc = {};
    for (int k0 = 0; k0 < K; k0 += 4) {
        const v2f a = *(const v2f*)(Arow + k0 + half * 2);  // global_load_b64
        const int q = k0 >> 2;
        const v2f b = *(const v2f*)(lds +                    // ds_load_b64
            (size_t)((((q << 1) + half) * ncols + col0 + m) << 1));
        acc = __builtin_amdgcn_wmma_f32_16x16x4_f32(false, a, false, b,
                                                    (short)0, acc, false, false);
    }

    const float bval = ldsBias[col0 + m];
#pragma unroll
    for (int r = 0; r < 8; ++r) {
        const long long mrow = row0 + r + half * 8;         // C/D layout: VGPR r -> M=r / r+8
        if (mrow < N) {
            float v = acc[r] + bval;
            if (doRelu) v = fmaxf(v, 0.0f);
            C[mrow * (long long)ncols + col0 + m] = v;
        }
    }
}

// ---------------------------------------------------------------------------
// Edge gather-scale-scatter: agg[dst] += hw[src] * norm[e]   (64 lanes/edge)
// ---------------------------------------------------------------------------
__global__ __launch_bounds__(256)
void scatter_kernel(const float* __restrict__ hw, const int* __restrict__ ei,
                    const float* __restrict__ nrm, float* __restrict__ agg, int E) {
    const long long g = (long long)blockIdx.x * blockDim.x + threadIdx.x;
    const long long e = g >> 6;
    const int f = (int)(g & 63);
    if (e >= E) return;
    // prefetch a future edge's gather row into cache (global_prefetch_b8)
    const long long e2 = e + 1024;
    if (e2 < E) {
        const long long s2 = ei[e2];
        __builtin_prefetch(&hw[s2 * 64 + f], 0, 1);
    }
    const long long s = ei[e];
    const long long d = ei[(long long)E + e];
    const float v = hw[s * 64 + f] * nrm[e];
    atomicAdd(&agg[d * 64 + f], v);
}

// ---------------------------------------------------------------------------
// out = relu(LN(agg + hw*self_norm + b_g)) (+ residual); one wave per node,
// 2 features per lane, wave32 shfl_xor reductions for mean/var.
// ---------------------------------------------------------------------------
__global__ __launch_bounds__(256)
void gcn_finish_kernel(const float* __restrict__ agg, const float* __restrict__ hw,
                       const float* __restrict__ selfn, const float* __restrict__ bg,
                       const float* __restrict__ residual, float* __restrict__ out, int N) {
    const int node = blockIdx.x * 8 + (threadIdx.x >> 5);
    const int lane = threadIdx.x & 31;
    if (node >= N) return;
    const long long base = (long long)node * 64;
    const float sn = selfn[node];
    float v0 = agg[base + lane]      + hw[base + lane]      * sn + bg[lane];
    float v1 = agg[base + lane + 32] + hw[base + lane + 32] * sn + bg[lane + 32];

    float s = v0 + v1;
    for (int off = 16; off; off >>= 1) s += __shfl_xor(s, off, 32);
    const float mean = s * (1.0f / 64.0f);
    const float d0 = v0 - mean, d1 = v1 - mean;
    float q = d0 * d0 + d1 * d1;
    for (int off = 16; off; off >>= 1) q += __shfl_xor(q, off, 32);
    const float inv = rsqrtf(q * (1.0f / 64.0f) + 1e-5f);

    float o0 = fmaxf(d0 * inv, 0.0f);
    float o1 = fmaxf(d1 * inv, 0.0f);
    if (residual) { o0 += residual[base + lane]; o1 += residual[base + lane + 32]; }
    out[base + lane] = o0;
    out[base + lane + 32] = o1;
}

// ---------------------------------------------------------------------------
// attn = sigmoid(t1 @ Wa2 + ba2); hattn = h * attn. One wave per node
// (lane l holds t1[node,l], l=0..31 exactly fills wave32).
// ---------------------------------------------------------------------------
__global__ __launch_bounds__(256)
void attn_kernel(const float* __restrict__ t1, const float* __restrict__ Wa2,
                 const float* __restrict__ ba2, const float* __restrict__ h,
                 float* __restrict__ hattn, float* __restrict__ attn_out, int N) {
    const int node = blockIdx.x * 8 + (threadIdx.x >> 5);
    const int lane = threadIdx.x & 31;
    if (node >= N) return;
    float p = t1[(long long)node * 32 + lane] * Wa2[lane];
    for (int off = 16; off; off >>= 1) p += __shfl_xor(p, off, 32);
    const float a = 1.0f / (1.0f + __expf(-(p + ba2[0])));
    if (lane == 0) attn_out[node] = a;
    const long long base = (long long)node * 64;
    hattn[base + lane]      = h[base + lane]      * a;
    hattn[base + lane + 32] = h[base + lane + 32] * a;
}

// ---------------------------------------------------------------------------
// logits[node,c] = t2[node,:] @ Wc2[:,c] + bc2[c]; one wave per node.
// ---------------------------------------------------------------------------
__global__ __launch_bounds__(256)
void logits_kernel(const float* __restrict__ t2, const float* __restrict__ Wc2,
                   const float* __restrict__ bc2, float* __restrict__ out, int N) {
    const int node = blockIdx.x * 8 + (threadIdx.x >> 5);
    const int lane = threadIdx.x & 31;
    if (node >= N) return;
    const float t = t2[(long long)node * 32 + lane];
    float s0 = t * Wc2[lane * 4 + 0];
    float s1 = t * Wc2[lane * 4 + 1];
    float s2 = t * Wc2[lane * 4 + 2];
    float s3 = t * Wc2[lane * 4 + 3];
    for (int off = 16; off; off >>= 1) {
        s0 += __shfl_xor(s0, off, 32);
        s1 += __shfl_xor(s1, off, 32);
        s2 += __shfl_xor(s2, off, 32);
        s3 += __shfl_xor(s3, off, 32);
    }
    if (lane == 0) {
        const long long o = (long long)node * 4;
        out[o + 0] = s0 + bc2[0];
        out[o + 1] = s1 + bc2[1];
        out[o + 2] = s2 + bc2[2];
        out[o + 3] = s3 + bc2[3];
    }
}

// ---------------------------------------------------------------------------
extern "C" void kernel_launch(void* const* d_in, const int* in_sizes, int n_in,
                              void* d_out, int out_size, void* d_ws, size_t ws_size,
                              hipStream_t stream) {
    const float* x   = (const float*)d_in[0];
    const float* ea  = (const float*)d_in[1];
    const int*   ei  = (const int*)  d_in[2];
    const float* We  = (const float*)d_in[3];
    const float* be  = (const float*)d_in[4];
    const float* Win = (const float*)d_in[5];
    const float* bin = (const float*)d_in[6];
    const float* Wg  = (const float*)d_in[7];
    const float* bg  = (const float*)d_in[8];
    const float* Wa1 = (const float*)d_in[9];
    const float* ba1 = (const float*)d_in[10];
    const float* Wa2 = (const float*)d_in[11];
    const float* ba2 = (const float*)d_in[12];
    const float* Wc1 = (const float*)d_in[13];
    const float* bc1 = (const float*)d_in[14];
    const float* Wc2 = (const float*)d_in[15];
    const float* bc2 = (const float*)d_in[16];

    const int N = in_sizes[0] / 128;
    const int E = in_sizes[1];
    const int H = 64, D = 128, Hh = 32;

    // workspace layout (floats)
    float* ws = (float*)d_ws;
    size_t o = 0;
    float* hA    = ws + o; o += (size_t)N * H;
    float* hB    = ws + o; o += (size_t)N * H;
    float* hw    = ws + o; o += (size_t)N * H;
    float* agg   = ws + o; o += (size_t)N * H;
    float* ew    = ws + o; o += (size_t)E;
    float* nrm   = ws + o; o += (size_t)E;
    float* deg   = ws + o; o += (size_t)N;
    float* dinv  = ws + o; o += (size_t)N;
    float* selfn = ws + o; o += (size_t)N;

    float* logits   = (float*)d_out;                  // [N,4]
    float* attn_out = (float*)d_out + (size_t)N * 4;  // [N,1]

    const int TB = 256;
    const int egrid = (E + TB - 1) / TB;
    const int ngrid = (N + TB - 1) / TB;
    const int wgrid = (N + 7) / 8;              // 8 waves (nodes) per block

    // --- edge weights, degrees, norms ---
    hipMemsetAsync(deg, 0, (size_t)N * sizeof(float), stream);
    ew_deg_kernel<<<egrid, TB, 0, stream>>>(ea, ei, We, be, ew, deg, E);
    dinv_kernel<<<ngrid, TB, 0, stream>>>(deg, dinv, selfn, N);
    norm_kernel<<<egrid, TB, 0, stream>>>(ei, ew, dinv, nrm, E);

    // --- input projection: h = relu(x @ W_in + b_in) ---
    {
        const int rows = 32;                                   // 8 waves / 4 col tiles
        const int grid = (N + rows - 1) / rows;
        const size_t sh = (size_t)(D * H + H) * sizeof(float); // 33 KB
        gemm_wmma_kernel<<<grid, TB, sh, stream>>>(x, Win, bin, hA, N, D, H, 1);
    }

    // --- 3 GCN layers ---
    float* hin = hA;
    float* hout = hB;
    const size_t sgrid = ((size_t)E * 64 + TB - 1) / TB;
    for (int i = 0; i < 3; ++i) {
        const int rows = 32;
        const int grid = (N + rows - 1) / rows;
        const size_t sh = (size_t)(H * H + H) * sizeof(float);
        gemm_wmma_kernel<<<grid, TB, sh, stream>>>(hin, Wg + (size_t)i * H * H,
                                                   nullptr, hw, N, H, H, 0);
        hipMemsetAsync(agg, 0, (size_t)N * H * sizeof(float), stream);
        scatter_kernel<<<(int)sgrid, TB, 0, stream>>>(hw, ei, nrm, agg, E);
        gcn_finish_kernel<<<wgrid, TB, 0, stream>>>(agg, hw, selfn, bg + (size_t)i * H,
                                                    (i > 0) ? hin : nullptr, hout, N);
        float* t = hin; hin = hout; hout = t;
    }
    // final node features in hin

    // --- attention head ---
    float* t1 = agg;     // [N,32], agg free now
    {
        const int rows = 64;                                   // 8 waves / 2 col tiles
        const int grid = (N + rows - 1) / rows;
        const size_t sh = (size_t)(H * Hh + Hh) * sizeof(float);
        gemm_wmma_kernel<<<grid, TB, sh, stream>>>(hin, Wa1, ba1, t1, N, H, Hh, 1);
    }
    float* hattn = hw;   // [N,64], hw free now
    attn_kernel<<<wgrid, TB, 0, stream>>>(t1, Wa2, ba2, hin, hattn, attn_out, N);

    // --- classifier ---
    float* t2 = hout;    // [N,32], spare h buffer
    {
        const int rows = 64;
        const int grid = (N + rows - 1) / rows;
        const size_t sh = (size_t)(H * Hh + Hh) * sizeof(float);
        gemm_wmma_kernel<<<grid, TB, sh, stream>>>(hattn, Wc1, bc1, t2, N, H, Hh, 1);
    }
    logits_kernel<<<wgrid, TB, 0, stream>>>(t2, Wc2, bc2, logits, N);
}